// MAM_52106543235451
// MI455X (gfx1250) — compile-verified
//
#include <hip/hip_runtime.h>

typedef float v2f __attribute__((ext_vector_type(2)));
typedef float v4f __attribute__((ext_vector_type(4)));
typedef float v8f __attribute__((ext_vector_type(8)));

#define BB 16
#define SS 4096
#define HH 12
#define DD 64
#define SCH 16              // S-chunks per (b,h)
#define ROWS (SS / SCH)     // 256 rows per chunk
#define NBH (BB * HH)       // 192

// ---------------------------------------------------------------------------
// Pass 1: column sums over S via V_WMMA_F32_16X16X4_F32 with all-ones A.
// One wave per (b,h,chunk). D[m,n] = sum_k B[k,n] -> row 0 of C = col sums.
// ---------------------------------------------------------------------------
__global__ __launch_bounds__(32) void k_colsum(const float* __restrict__ x,
                                               float* __restrict__ partial) {
    const int bh    = blockIdx.x;          // 0..191
    const int chunk = blockIdx.y;          // 0..15
    const int b = bh / HH, h = bh % HH;
    const int lane = threadIdx.x;
    const int half = lane >> 4;            // 0: lanes 0-15, 1: lanes 16-31
    const int n    = lane & 15;            // column within 16-wide tile

    // lane base: rows s0 + 2*half (+1 for second VGPR slot), col d = n
    const float* base =
        x + ((size_t)(b * SS + chunk * ROWS + 2 * half) * HH + h) * DD + n;

    v2f a; a.x = 1.0f; a.y = 1.0f;          // all-ones 16x4 A matrix
    v8f c0 = {}, c1 = {}, c2 = {}, c3 = {}; // four 16-col tiles -> D = 64

    for (int k4 = 0; k4 < ROWS / 4; ++k4) { // 4 s-rows per WMMA step
        const float* p = base + (size_t)k4 * 4 * (HH * DD);
        v2f b0, b1, b2, b3;
        b0.x = p[0];  b0.y = p[HH * DD];
        b1.x = p[16]; b1.y = p[HH * DD + 16];
        b2.x = p[32]; b2.y = p[HH * DD + 32];
        b3.x = p[48]; b3.y = p[HH * DD + 48];
        c0 = __builtin_amdgcn_wmma_f32_16x16x4_f32(false, a, false, b0, (short)0, c0, false, false);
        c1 = __builtin_amdgcn_wmma_f32_16x16x4_f32(false, a, false, b1, (short)0, c1, false, false);
        c2 = __builtin_amdgcn_wmma_f32_16x16x4_f32(false, a, false, b2, (short)0, c2, false, false);
        c3 = __builtin_amdgcn_wmma_f32_16x16x4_f32(false, a, false, b3, (short)0, c3, false, false);
    }

    if (lane < 16) {                        // C row M=0 lives in VGPR0, lanes 0-15
        float* o = partial + ((size_t)bh * SCH + chunk) * DD;
        o[n]      = c0[0];
        o[16 + n] = c1[0];
        o[32 + n] = c2[0];
        o[48 + n] = c3[0];
    }
}

// ---------------------------------------------------------------------------
// Pass 2: deterministic chunk reduction, head/batch means, top-k selection,
// precompute rep[bh][d] = scale * col_sum and mode flags. Single block.
// ---------------------------------------------------------------------------
__global__ __launch_bounds__(1024) void k_select(const float* __restrict__ partial,
                                                 const float* __restrict__ hparam,
                                                 float* __restrict__ rep,
                                                 int* __restrict__ mode) {
    __shared__ float ls_col[NBH * DD];   // 12288 floats = 48KB
    __shared__ float ls_hm[NBH];
    __shared__ float ls_scale[NBH];
    __shared__ int   ls_mode[NBH];
    const int tid = threadIdx.x;

    // fixed-order reduction over the 16 S-chunks
    for (int i = tid; i < NBH * DD; i += 1024) {
        const int bh = i >> 6, d = i & 63;
        const float* p = partial + (size_t)bh * (SCH * DD) + d;
        float s = 0.0f;
        #pragma unroll
        for (int c = 0; c < SCH; ++c) s += p[c * DD];
        ls_col[i] = s;
    }
    __syncthreads();

    if (tid < NBH) {
        float hs = 0.0f;
        for (int d = 0; d < DD; ++d) hs += ls_col[tid * DD + d];
        ls_hm[tid] = hs * (1.0f / ((float)SS * (float)DD));   // head_mean
    }
    __syncthreads();

    if (tid < BB) {
        float hm[HH], dev[HH];
        float bm = 0.0f;
        for (int h = 0; h < HH; ++h) { hm[h] = ls_hm[tid * HH + h]; bm += hm[h]; }
        bm *= (1.0f / HH);                                    // batch_mean
        for (int h = 0; h < HH; ++h) dev[h] = fabsf(hm[h] - bm);

        float hw = hparam[0];
        hw = fminf(fmaxf(hw, 5.0f), 10.0f);
        const float sHigh = hw * 0.2f;
        const float sLow  = (10.0f - hw) * 0.2f;

        for (int h = 0; h < HH; ++h) {
            int cmax = 0, cmin = 0;
            for (int j = 0; j < HH; ++j) {   // lax.top_k tie-break: lower index first
                if (dev[j] > dev[h] || (dev[j] == dev[h] && j < h)) ++cmax;
                if (dev[j] < dev[h] || (dev[j] == dev[h] && j < h)) ++cmin;
            }
            const bool isMax = (cmax < 3);
            const bool isMin = (cmin < 3) && !isMax;          // max mask wins overlap
            ls_scale[tid * HH + h] = isMax ? sHigh : (isMin ? sLow : 0.0f);
            ls_mode[tid * HH + h]  = (isMax || isMin) ? 1 : 0;
        }
    }
    __syncthreads();

    for (int i = tid; i < NBH * DD; i += 1024)
        rep[i] = ls_scale[i >> 6] * ls_col[i];
    if (tid < NBH) mode[tid] = ls_mode[tid];
}

// ---------------------------------------------------------------------------
// Pass 3: streaming rewrite, float4 per thread, non-temporal hints.
// ---------------------------------------------------------------------------
__global__ __launch_bounds__(256) void k_write(const float* __restrict__ x,
                                               const float* __restrict__ rep,
                                               const int* __restrict__ mode,
                                               float* __restrict__ out) {
    const int idx4 = blockIdx.x * 256 + threadIdx.x;      // < 12,582,912
    const int h  = (idx4 >> 4) % HH;                      // 16 float4s per head-row
    const int b  = idx4 / (SS * HH * DD / 4);             // 786432 float4s per batch
    const int bh = b * HH + h;

    const v4f* x4 = (const v4f*)x;
    const v4f* r4 = (const v4f*)rep;
    v4f*       o4 = (v4f*)out;

    v4f v;
    if (mode[bh]) v = r4[bh * 16 + (idx4 & 15)];          // hot in L2 (48KB)
    else          v = __builtin_nontemporal_load(x4 + idx4);
    __builtin_nontemporal_store(v, o4 + idx4);
}

// ---------------------------------------------------------------------------
extern "C" void kernel_launch(void* const* d_in, const int* in_sizes, int n_in,
                              void* d_out, int out_size, void* d_ws, size_t ws_size,
                              hipStream_t stream) {
    const float* x  = (const float*)d_in[0];
    const float* hp = (const float*)d_in[1];
    float* out = (float*)d_out;

    float* partial = (float*)d_ws;                 // 192*16*64 floats = 768KB
    float* rep     = partial + NBH * SCH * DD;     // 12288 floats
    int*   mode    = (int*)(rep + NBH * DD);       // 192 ints

    k_colsum<<<dim3(NBH, SCH), 32, 0, stream>>>(x, partial);
    k_select<<<1, 1024, 0, stream>>>(partial, hp, rep, mode);

    const int n4 = BB * SS * HH * DD / 4;          // 12,582,912
    k_write<<<n4 / 256, 256, 0, stream>>>(x, rep, mode, out);
}